// CM_Module_11330123727516
// MI455X (gfx1250) — compile-verified
//
#include <hip/hip_runtime.h>
#include <hip/hip_bf16.h>
#include <cmath>

// ---------------- problem constants (from reference) ----------------
#define T_   8
#define B_   4
#define C_   128
#define Hc   96
#define Wc   96
#define HWQ  (Hc*Wc)            // 9216
#define CHW  (C_*HWQ)           // 1,179,648
#define BCHW (B_*CHW)           // 4,718,592
#define HIRES (384*384)         // 147,456
#define NBLK 36                 // 9216 / 256 pixel-blocks

typedef __attribute__((ext_vector_type(2))) float v2f;
typedef __attribute__((ext_vector_type(8))) float v8f;

// Exact f32 wave(32) sum using the matrix unit:
// A(16x4): lane m vgpr0 = acc[m] (K=0), lanes 16..31 vgpr0 = acc[m+16] (K=2),
// vgpr1 = 0 (K=1,3). B = ones. D[m,n] = acc[m] + acc[m+16] for every n.
// Lanes 0..15 hold rows M=0..7 in d[0..7]; lanes 16..31 hold rows M=8..15.
// So per-lane sum of d[0..7] + shfl_xor(16) = exact full-wave sum.
__device__ __forceinline__ float wave_reduce_sum(float x) {
#if __has_builtin(__builtin_amdgcn_wmma_f32_16x16x4_f32)
    v2f a; a.x = x;    a.y = 0.0f;
    v2f b; b.x = 1.0f; b.y = 1.0f;
    v8f c = {};
    c = __builtin_amdgcn_wmma_f32_16x16x4_f32(false, a, false, b,
                                              (short)0, c, false, false);
    float s = ((c[0] + c[1]) + (c[2] + c[3])) + ((c[4] + c[5]) + (c[6] + c[7]));
    s += __shfl_xor(s, 16, 32);
    return s;
#else
    for (int off = 16; off > 0; off >>= 1) x += __shfl_xor(x, off, 32);
    return x;
#endif
}

// ---------------- kernel 1: antialiased 4x bilinear downsample + binarize ---
// jax.image.resize(antialias=True): sample at 4*o+1.5, triangle kernel width 4,
// unnormalized taps {1,3,5,7,7,5,3,1}/32 at i = 4o-2..4o+5, renormalized over
// in-range taps per axis. Then (val > 0.5) ? 1 : 0.
__global__ void k_resize_bin(const float* __restrict__ tvmap,
                             const float* __restrict__ rvmaps,
                             float* __restrict__ tv,
                             float* __restrict__ rv) {
    const int m = blockIdx.y;                        // 0..35 (4 tv + 32 rv maps)
    const int p = blockIdx.x * 256 + threadIdx.x;    // 0..9215
    const int oy = p / Wc, ox = p % Wc;
    const float WT[8] = {1.f/32, 3.f/32, 5.f/32, 7.f/32,
                         7.f/32, 5.f/32, 3.f/32, 1.f/32};
    const float* src;
    float* dst;
    if (m < B_) { src = tvmap  + (size_t)m * HIRES;       dst = tv + (size_t)m * HWQ; }
    else        { src = rvmaps + (size_t)(m-B_) * HIRES;  dst = rv + (size_t)(m-B_) * HWQ; }

    float wx[8]; int ixv[8]; float sx = 0.f;
#pragma unroll
    for (int k = 0; k < 8; ++k) {
        int ix = 4*ox - 2 + k;
        bool v = (unsigned)ix < 384u;
        float w = v ? WT[k] : 0.f;
        wx[k] = w; ixv[k] = v ? ix : 0; sx += w;
    }
    float num = 0.f, sy = 0.f;
#pragma unroll
    for (int j = 0; j < 8; ++j) {
        int iy = 4*oy - 2 + j;
        if ((unsigned)iy < 384u) {
            float wyj = WT[j]; sy += wyj;
            const float* row = src + (size_t)iy * 384;
            float r = 0.f;
#pragma unroll
            for (int k = 0; k < 8; ++k) r = fmaf(wx[k], row[ixv[k]], r);
            num = fmaf(wyj, r, num);
        }
    }
    float val = num / (sy * sx);
    dst[p] = (val > 0.5f) ? 1.0f : 0.0f;
}

// ---------------- kernel 2: per-block partials of gs / v_sum ----------------
// grid = (NBLK, 32); blockIdx.y = q = t*B + b; each thread owns one pixel.
__global__ void k_gs_partial(const float* __restrict__ values,
                             const float* __restrict__ tv,
                             const float* __restrict__ rv,
                             float* __restrict__ pg,
                             float* __restrict__ pv) {
    const int q = blockIdx.y;
    const int t = q >> 2, b = q & 3;
    const int p = blockIdx.x * 256 + threadIdx.x;

    const float mask = tv[b*HWQ + p] * rv[q*HWQ + p];   // {0,1}
    float acc = 0.f;
    if (mask != 0.f) {
        const float* tf = values + (size_t)b * CHW + p;                       // t_feat (L2-resident)
        const float* rf = values + (size_t)(1 + t) * BCHW + (size_t)b * CHW + p;
#pragma unroll 8
        for (int c = 0; c < C_; ++c)
            acc = fmaf(tf[(size_t)c * HWQ], rf[(size_t)c * HWQ], acc);
    }
    // whole block is converged here -> EXEC all ones, legal for WMMA
    const float gsum = wave_reduce_sum(acc);
    const float vsum = wave_reduce_sum(mask);

    __shared__ float sg[8], sv[8];
    const int wid = threadIdx.x >> 5, lane = threadIdx.x & 31;
    if (lane == 0) { sg[wid] = gsum; sv[wid] = vsum; }
    __syncthreads();
    if (threadIdx.x == 0) {
        float G = 0.f, V = 0.f;
        for (int i = 0; i < 8; ++i) { G += sg[i]; V += sv[i]; }
        pg[q*NBLK + blockIdx.x] = G;
        pv[q*NBLK + blockIdx.x] = V;
    }
}

// ---------------- kernel 3: finalize gs (deterministic fixed-order sum) -----
__global__ void k_gs_final(const float* __restrict__ pg,
                           const float* __restrict__ pv,
                           float* __restrict__ gsf,        // ws copy for kernel 4
                           float* __restrict__ gs_out) {   // d_out slice (T,B)
    const int q = threadIdx.x;
    if (q >= 32) return;
    float G = 0.f, V = 0.f;
    for (int i = 0; i < NBLK; ++i) { G += pg[q*NBLK + i]; V += pv[q*NBLK + i]; }
    const bool empty = V < 1e-4f;
    float g = empty ? 0.f : G;
    g = g / (V + (empty ? 1.f : 0.f));
    g = g * (1.0f / (float)C_);          // /C, exact (power of two)
    gsf[q] = g;
    gs_out[q] = g;
}

// ---------------- kernel 4: per-pixel weights + outputs ---------------------
// grid = (NBLK, B); one thread per (b, pixel).
__global__ void k_output(const float* __restrict__ values,
                         const float* __restrict__ rv,
                         const float* __restrict__ gsf,
                         float* __restrict__ out,        // (B, 2C+1, H, W)
                         float* __restrict__ cmask_out)  // (B, 1, H, W)
{
    const int b = blockIdx.y;
    const int p = blockIdx.x * 256 + threadIdx.x;

    float rvt[T_], mv[T_];
#pragma unroll
    for (int t = 0; t < T_; ++t) {
        rvt[t] = rv[(t*B_ + b)*HWQ + p];
        mv[t]  = gsf[t*B_ + b] * rvt[t];
    }
    float mx = mv[0];
#pragma unroll
    for (int t = 1; t < T_; ++t) mx = fmaxf(mx, mv[t]);
    float me[T_], ms = 0.f;
#pragma unroll
    for (int t = 0; t < T_; ++t) { me[t] = expf(mv[t] - mx) * rvt[t]; ms += me[t]; }
    if (ms < 1e-4f) ms += 1.0f;
    const float inv = 1.0f / ms;
    float cm[T_], cmsum = 0.f;
#pragma unroll
    for (int t = 0; t < T_; ++t) { cm[t] = me[t] * inv; cmsum = fmaf(cm[t], rvt[t], cmsum); }

    const float cmask = 1.0f - cmsum;
    float* outb = out + (size_t)b * (2*C_ + 1) * HWQ;
    outb[(size_t)(2*C_) * HWQ + p] = cmask;
    cmask_out[(size_t)b * HWQ + p] = cmask;

    const float* tf  = values + (size_t)b * CHW + p;          // t_feat
    const float* rfb = values + (size_t)BCHW + (size_t)b * CHW + p; // r_feats t=0
    for (int c = 0; c < C_; ++c) {
        if (c + 4 < C_) __builtin_prefetch(rfb + (size_t)(c + 4) * HWQ, 0, 0);
        outb[(size_t)c * HWQ + p] = tf[(size_t)c * HWQ];      // passthrough t_feat
        float acc = 0.f;
#pragma unroll
        for (int t = 0; t < T_; ++t)
            acc = fmaf(rfb[(size_t)t * BCHW + (size_t)c * HWQ], cm[t], acc);
        outb[(size_t)(C_ + c) * HWQ + p] = acc;               // c_out
    }
}

// ---------------- launch ----------------------------------------------------
extern "C" void kernel_launch(void* const* d_in, const int* in_sizes, int n_in,
                              void* d_out, int out_size, void* d_ws, size_t ws_size,
                              hipStream_t stream) {
    const float* values = (const float*)d_in[0];   // (9,4,128,96,96)
    const float* tvmap  = (const float*)d_in[1];   // (4,1,384,384)
    const float* rvmaps = (const float*)d_in[2];   // (8,4,1,384,384)
    float* out = (float*)d_out;

    // workspace layout (floats): needs ~1.34 MB
    float* ws_tv  = (float*)d_ws;            // 4*9216
    float* ws_rv  = ws_tv + B_*HWQ;          // 32*9216
    float* ws_pg  = ws_rv + T_*B_*HWQ;       // 32*36
    float* ws_pv  = ws_pg + 32*NBLK;         // 32*36
    float* ws_gsf = ws_pv + 32*NBLK;         // 32

    // d_out layout: out (4*257*9216) | c_mask (4*9216) | gs (32)
    float* out_cmask = out + (size_t)B_ * (2*C_ + 1) * HWQ;
    float* out_gs    = out_cmask + (size_t)B_ * HWQ;

    dim3 blk(256);
    // 1) downsample + binarize the 36 masks
    k_resize_bin<<<dim3(NBLK, B_ + T_*B_), blk, 0, stream>>>(tvmap, rvmaps, ws_tv, ws_rv);
    // 2) gs / v_sum block partials (WMMA wave reductions)
    k_gs_partial<<<dim3(NBLK, T_*B_), blk, 0, stream>>>(values, ws_tv, ws_rv, ws_pg, ws_pv);
    // 3) finalize gs
    k_gs_final<<<dim3(1), dim3(32), 0, stream>>>(ws_pg, ws_pv, ws_gsf, out_gs);
    // 4) softmax-matched aggregation + outputs
    k_output<<<dim3(NBLK, B_), blk, 0, stream>>>(values, ws_rv, ws_gsf, out, out_cmask);
}